// TSPDDecoder_31542239822220
// MI455X (gfx1250) — compile-verified
//
#include <hip/hip_runtime.h>
#include <hip/hip_bf16.h>
#include <math.h>

typedef float v2f __attribute__((ext_vector_type(2)));
typedef float v8f __attribute__((ext_vector_type(8)));

#define BB 512
#define NN 1000
#define DD 128
#define DYN 4
#define KTOT 132          // 128 static dims + 4 dynamic-feature dims
#define KPAD 134          // LDS row stride (words): 134%64=6 -> conflict-free b64 frags

// workspace layout (float offsets)
#define WS_WT 0                    // W'^T  [256][132]
#define WS_BT (256 * KTOT)         // truck fused bias  [128]
#define WS_BD (WS_BT + 128)        // drone fused bias  [128]
#define WS_Q  (WS_BD + 128)        // per-batch query vec [512][256]

// ---------------------------------------------------------------------------
// k1: build fused transposed weight W'^T and fused biases.
//   W'[k][j] (k<128):  j<128 ? tWs[k][j] : dWs[k][j-128]
//   W'[128+c][j]:      (W_embed @ Wd)[c][j]        (rank-4 E_dynamic path)
//   bias[j] = b_embed @ Wd[:,j] + bd[j] + bs[j]
// ---------------------------------------------------------------------------
__global__ void k1_prep(const float* __restrict__ W_embed, const float* __restrict__ b_embed,
                        const float* __restrict__ tWd, const float* __restrict__ tbd,
                        const float* __restrict__ tWs, const float* __restrict__ tbs,
                        const float* __restrict__ dWd, const float* __restrict__ dbd,
                        const float* __restrict__ dWs, const float* __restrict__ dbs,
                        float* __restrict__ ws) {
  int j = blockIdx.x;                 // 0..255 output column
  int t = threadIdx.x;                // 0..127
  const float* Ws = (j < 128) ? tWs : dWs;
  const float* Wd = (j < 128) ? tWd : dWd;
  int jc = (j < 128) ? j : (j - 128);
  float* WT = ws + WS_WT;
  WT[j * KTOT + t] = Ws[t * 128 + jc];          // transpose copy, k = t
  if (t < 4) {                                   // rank-4 rows
    float acc = 0.f;
    for (int m = 0; m < 128; ++m) acc += W_embed[t * 128 + m] * Wd[m * 128 + jc];
    WT[j * KTOT + 128 + t] = acc;
  }
  if (t == 4) {                                  // fused bias
    float acc = 0.f;
    for (int m = 0; m < 128; ++m) acc += b_embed[m] * Wd[m * 128 + jc];
    if (j < 128) ws[WS_BT + jc] = acc + tbd[jc] + tbs[jc];
    else         ws[WS_BD + jc] = acc + dbd[jc] + dbs[jc];
  }
}

// ---------------------------------------------------------------------------
// k2: per-batch decoder path (ctx gather, dyn mean, input proj, 2x MGU,
//     h_new output, per-batch query vectors into workspace).
// one block per batch, 128 threads (one per feature d).
// ---------------------------------------------------------------------------
__global__ void __launch_bounds__(128)
k2_decode(const float* __restrict__ Es, const float* __restrict__ dyn,
          const float* __restrict__ h_prev,
          const int* __restrict__ t_loc, const int* __restrict__ d_loc,
          const float* __restrict__ W_embed, const float* __restrict__ b_embed,
          const float* __restrict__ W_in, const float* __restrict__ b_in,
          const float* __restrict__ Wf0, const float* __restrict__ bf0,
          const float* __restrict__ Wh0, const float* __restrict__ bh0,
          const float* __restrict__ Wf1, const float* __restrict__ bf1,
          const float* __restrict__ Wh1, const float* __restrict__ bh1,
          const float* __restrict__ tWq, const float* __restrict__ tbq,
          const float* __restrict__ dWq, const float* __restrict__ dbq,
          const float* __restrict__ ws, float* __restrict__ ws_q,
          float* __restrict__ out_h) {
  __shared__ float sIn[256];
  __shared__ float sH[128];
  __shared__ float sX[128];
  __shared__ float sF[128];
  __shared__ float red[128];
  __shared__ float sGm[4];
  int b = blockIdx.x, d = threadIdx.x;

  int tl = t_loc[b], dl = d_loc[b];
  sIn[d] = 0.5f * (Es[((size_t)b * NN + tl) * DD + d] + Es[((size_t)b * NN + dl) * DD + d]);

  { // mean over N of the 4 dynamic features
    int c = d & 3, g = d >> 2;
    float s = 0.f;
    for (int n = g; n < NN; n += 32) s += dyn[((size_t)b * NN + n) * DYN + c];
    red[d] = s;
  }
  __syncthreads();
  if (d < 4) {
    float s = 0.f;
    for (int g = 0; g < 32; ++g) s += red[g * 4 + d];
    sGm[d] = s * (1.0f / NN);
  }
  __syncthreads();
  sIn[128 + d] = sGm[0] * W_embed[0 * 128 + d] + sGm[1] * W_embed[1 * 128 + d] +
                 sGm[2] * W_embed[2 * 128 + d] + sGm[3] * W_embed[3 * 128 + d] + b_embed[d];
  sH[d] = h_prev[(b * 2 + 0) * 128 + d];
  __syncthreads();

  // dec_input = [ctx, gdyn] @ W_in + b_in
  float x = b_in[d];
  for (int k = 0; k < 256; ++k) x += sIn[k] * W_in[k * 128 + d];
  sX[d] = x;
  __syncthreads();

  for (int layer = 0; layer < 2; ++layer) {
    const float* Wf_ = layer ? Wf1 : Wf0;
    const float* bf_ = layer ? bf1 : bf0;
    const float* Wh_ = layer ? Wh1 : Wh0;
    const float* bh_ = layer ? bh1 : bh0;
    float a = bf_[d];
    for (int k = 0; k < 128; ++k) a += sH[k] * Wf_[k * 128 + d];
    for (int k = 0; k < 128; ++k) a += sX[k] * Wf_[(128 + k) * 128 + d];
    float f = 1.f / (1.f + expf(-a));
    sF[d] = f;
    __syncthreads();
    float g = bh_[d];
    for (int k = 0; k < 128; ++k) g += sF[k] * sH[k] * Wh_[k * 128 + d];
    for (int k = 0; k < 128; ++k) g += sX[k] * Wh_[(128 + k) * 128 + d];
    float ht = tanhf(g);
    float hnew = (1.f - f) * sH[d] + f * ht;
    out_h[(size_t)4 * BB * NN + (b * 2 + layer) * 128 + d] = hnew;
    __syncthreads();
    sX[d] = hnew;
    if (layer == 0) sH[d] = h_prev[(b * 2 + 1) * 128 + d];
    __syncthreads();
  }

  // query vectors: q[j] = h1 @ Wq[:,j] + bq[j] + fused_bias[j]
  float qt = tbq[d] + ws[WS_BT + d];
  float qd = dbq[d] + ws[WS_BD + d];
  for (int k = 0; k < 128; ++k) {
    float h = sX[k];
    qt += h * tWq[k * 128 + d];
    qd += h * dWq[k * 128 + d];
  }
  ws_q[b * 256 + d] = qt;
  ws_q[b * 256 + 128 + d] = qd;
}

// ---------------------------------------------------------------------------
// k3: the heavy kernel. One block (8 wave32) per batch.
//   Software-pipelined: strip i+1 is fetched from global into registers while
//   WMMA runs on strip i in LDS; LDS store happens after the epilogue barrier.
// ---------------------------------------------------------------------------
__device__ __forceinline__ void softmax_head(const float* __restrict__ lg,
                                             float* __restrict__ red,
                                             float* __restrict__ outL,
                                             float* __restrict__ outP, int tid) {
  float m = -INFINITY;
  for (int i = tid; i < NN; i += 256) m = fmaxf(m, lg[i]);
  red[tid] = m; __syncthreads();
  for (int s = 128; s > 0; s >>= 1) { if (tid < s) red[tid] = fmaxf(red[tid], red[tid + s]); __syncthreads(); }
  m = red[0]; __syncthreads();
  float ssum = 0.f;
  for (int i = tid; i < NN; i += 256) ssum += expf(lg[i] - m);
  red[tid] = ssum; __syncthreads();
  for (int s = 128; s > 0; s >>= 1) { if (tid < s) red[tid] += red[tid + s]; __syncthreads(); }
  float inv = 1.f / red[0]; __syncthreads();
  for (int i = tid; i < NN; i += 256) {
    float l = lg[i];
    outL[i] = l;
    outP[i] = expf(l - m) * inv;
  }
}

__global__ void __launch_bounds__(256)
k3_attn(const float* __restrict__ Es, const float* __restrict__ dyn,
        const unsigned char* __restrict__ maskT, const unsigned char* __restrict__ maskD,
        const float* __restrict__ tv, const float* __restrict__ tCp,
        const float* __restrict__ dv, const float* __restrict__ dCp,
        const float* __restrict__ ws, float* __restrict__ out) {
  __shared__ float Wt[256 * KPAD];   // W'^T staged, padded rows
  __shared__ float As[16 * KPAD];    // A strip [16 rows][132] padded
  __shared__ float sLogT[NN];
  __shared__ float sLogD[NN];
  __shared__ float uArr[32];
  __shared__ float red[256];

  int b = blockIdx.x;
  int tid = threadIdx.x;

  for (int idx = tid; idx < 256 * KTOT; idx += 256) {
    int j = idx / KTOT, k = idx - j * KTOT;
    Wt[j * KPAD + k] = ws[WS_WT + idx];
  }

  int lane = tid & 31, wave = tid >> 5;
  int half = lane >> 4, nl = lane & 15;
  int ko = 2 * half;                       // K split: lanes 0-15 -> K,K+1 ; 16-31 -> K+2,K+3
  int j0 = wave * 16 + nl;                 // this lane's output column within a head
  float qv0 = ws[WS_Q + b * 256 + j0];
  float qv1 = ws[WS_Q + b * 256 + 128 + j0];
  float vj0 = tv[j0];
  float vj1 = dv[j0];
  float tC = tCp[0], dC = dCp[0];

  int r = tid >> 4, cb = tid & 15;         // staging coords: row / column lane

  { // stage strip 0 (full KPAD width so pad cols 132..133 are zeroed once)
    const float* esrow = Es + ((size_t)b * NN + r) * DD;
    const float* dyrow = dyn + ((size_t)b * NN + r) * DYN;
    for (int c = cb; c < KPAD; c += 16) {
      float v = 0.f;
      if (c < KTOT) v = (c < 128) ? esrow[c] : dyrow[c - 128];
      As[r * KPAD + c] = v;
    }
  }
  if (tid < 32) uArr[tid] = 0.f;
  __syncthreads();

  for (int n0 = 0; n0 < NN; n0 += 16) {
    // ---- register prefetch of strip n0+16 (overlaps with WMMA below) ----
    float pre[9];
    bool have_next = (n0 + 16 < NN);
    {
      int n1 = n0 + 16 + r;
      const float* esrow = Es + ((size_t)b * NN + n1) * DD;
      const float* dyrow = dyn + ((size_t)b * NN + n1) * DYN;
      bool rv = have_next && (n1 < NN);
#pragma unroll
      for (int i = 0; i < 9; ++i) {
        int c = cb + 16 * i;
        float v = 0.f;
        if (rv && c < KTOT) v = (c < 128) ? esrow[c] : dyrow[c - 128];
        pre[i] = v;
      }
    }
    // L2/WGP$ prefetch for strip n0+32 (4 x 128B lines per 512B row)
    if (tid < 64) {
      int n2 = n0 + 32 + (tid >> 2);
      if (n2 < NN)
        __builtin_prefetch(Es + ((size_t)b * NN + n2) * DD + (tid & 3) * 32, 0, 3);
    }

    // ---- WMMA GEMM on current strip ----
    v8f c0 = {0.f, 0.f, 0.f, 0.f, 0.f, 0.f, 0.f, 0.f};
    v8f c1 = c0;
    for (int k = 0; k < KTOT; k += 4) {
      v2f a  = *(const v2f*)&As[nl * KPAD + k + ko];                     // A: M=nl, K pair
      v2f b0 = *(const v2f*)&Wt[(wave * 16 + nl) * KPAD + k + ko];       // B truck: N=nl
      v2f b1 = *(const v2f*)&Wt[(128 + wave * 16 + nl) * KPAD + k + ko]; // B drone
      c0 = __builtin_amdgcn_wmma_f32_16x16x4_f32(false, a, false, b0, (short)0, c0, false, false);
      c1 = __builtin_amdgcn_wmma_f32_16x16x4_f32(false, a, false, b1, (short)0, c1, false, false);
    }

    // ---- fused epilogue: tanh(s + q) * v_j, reduce 16 cols, accumulate u[n] ----
#pragma unroll
    for (int v = 0; v < 8; ++v) {
      float t0 = tanhf(c0[v] + qv0) * vj0;
      float t1 = tanhf(c1[v] + qv1) * vj1;
#pragma unroll
      for (int m = 1; m < 16; m <<= 1) {
        t0 += __shfl_xor(t0, m);
        t1 += __shfl_xor(t1, m);
      }
      if (nl == 0) {                       // lane 0 -> rows M=v ; lane 16 -> rows M=v+8
        atomicAdd(&uArr[v + 8 * half], t0);
        atomicAdd(&uArr[16 + v + 8 * half], t1);
      }
    }
    __syncthreads();   // u complete; all LDS A reads done

    if (tid < 16) {
      int n = n0 + tid;
      if (n < NN) {
        float lt = tC * tanhf(uArr[tid]);
        if (maskT[(size_t)b * NN + n]) lt = -INFINITY;
        sLogT[n] = lt;
        float ld = dC * tanhf(uArr[16 + tid]);
        if (maskD[(size_t)b * NN + n]) ld = -INFINITY;
        sLogD[n] = ld;
      }
      uArr[tid] = 0.f;           // re-arm accumulators for next strip
      uArr[16 + tid] = 0.f;
    }
    if (have_next) {             // commit prefetched strip into LDS
#pragma unroll
      for (int i = 0; i < 9; ++i) {
        int c = cb + 16 * i;
        if (c < KTOT) As[r * KPAD + c] = pre[i];
      }
    }
    __syncthreads();
  }

  float* outTL = out;
  float* outTP = out + (size_t)BB * NN;
  float* outDL = out + (size_t)2 * BB * NN;
  float* outDP = out + (size_t)3 * BB * NN;
  softmax_head(sLogT, red, outTL + (size_t)b * NN, outTP + (size_t)b * NN, tid);
  __syncthreads();
  softmax_head(sLogD, red, outDL + (size_t)b * NN, outDP + (size_t)b * NN, tid);
}

// ---------------------------------------------------------------------------
extern "C" void kernel_launch(void* const* d_in, const int* in_sizes, int n_in,
                              void* d_out, int out_size, void* d_ws, size_t ws_size,
                              hipStream_t stream) {
  (void)in_sizes; (void)n_in; (void)out_size; (void)ws_size;
  const float* Es      = (const float*)d_in[0];
  const float* dyn     = (const float*)d_in[1];
  const float* h_prev  = (const float*)d_in[2];
  const int*   t_loc   = (const int*)d_in[3];
  const int*   d_loc   = (const int*)d_in[4];
  const unsigned char* mT = (const unsigned char*)d_in[5];
  const unsigned char* mD = (const unsigned char*)d_in[6];
  const float* W_embed = (const float*)d_in[7];
  const float* b_embed = (const float*)d_in[8];
  const float* W_in    = (const float*)d_in[9];
  const float* b_in    = (const float*)d_in[10];
  const float* Wf0 = (const float*)d_in[11];
  const float* bf0 = (const float*)d_in[12];
  const float* Wh0 = (const float*)d_in[13];
  const float* bh0 = (const float*)d_in[14];
  const float* Wf1 = (const float*)d_in[15];
  const float* bf1 = (const float*)d_in[16];
  const float* Wh1 = (const float*)d_in[17];
  const float* bh1 = (const float*)d_in[18];
  const float* tWd = (const float*)d_in[19];
  const float* tbd = (const float*)d_in[20];
  const float* tWs = (const float*)d_in[21];
  const float* tbs = (const float*)d_in[22];
  const float* tWq = (const float*)d_in[23];
  const float* tbq = (const float*)d_in[24];
  const float* tv  = (const float*)d_in[25];
  const float* tC  = (const float*)d_in[26];
  const float* dWd = (const float*)d_in[27];
  const float* dbd = (const float*)d_in[28];
  const float* dWs = (const float*)d_in[29];
  const float* dbs = (const float*)d_in[30];
  const float* dWq = (const float*)d_in[31];
  const float* dbq = (const float*)d_in[32];
  const float* dv  = (const float*)d_in[33];
  const float* dC  = (const float*)d_in[34];

  float* ws  = (float*)d_ws;
  float* out = (float*)d_out;

  k1_prep<<<256, 128, 0, stream>>>(W_embed, b_embed, tWd, tbd, tWs, tbs,
                                   dWd, dbd, dWs, dbs, ws);
  k2_decode<<<BB, 128, 0, stream>>>(Es, dyn, h_prev, t_loc, d_loc, W_embed, b_embed,
                                    W_in, b_in, Wf0, bf0, Wh0, bh0, Wf1, bf1, Wh1, bh1,
                                    tWq, tbq, dWq, dbq, ws, ws + WS_Q, out);
  k3_attn<<<BB, 256, 0, stream>>>(Es, dyn, mT, mD, tv, tC, dv, dC, ws, out);
}